// MMGCN_36464272343526
// MI455X (gfx1250) — compile-verified
//
#include <hip/hip_runtime.h>
#include <hip/hip_bf16.h>
#include <math.h>

typedef __attribute__((ext_vector_type(2))) float v2f;
typedef __attribute__((ext_vector_type(8))) float v8f;
typedef __attribute__((address_space(3))) float lds_f32;

#define SLOPE 0.01f
#define EPSN  1e-12f

__device__ __forceinline__ float lrelu(float x) {
    return x >= 0.0f ? x : SLOPE * x;
}

// ---------------------------------------------------------------------------
// Generic zero-fill (grid-stride)
// ---------------------------------------------------------------------------
__global__ void zero_f32(float* __restrict__ p, long long n) {
    long long i = (long long)blockIdx.x * blockDim.x + threadIdx.x;
    long long stride = (long long)gridDim.x * blockDim.x;
    for (; i < n; i += stride) p[i] = 0.0f;
}

// ---------------------------------------------------------------------------
// Word-embedding scatter: sums[item] += word_emb[word]; cnt[item] += 1
// ---------------------------------------------------------------------------
__global__ void word_scatter(const float* __restrict__ word_emb,
                             const int* __restrict__ item_ids,
                             const int* __restrict__ word_ids,
                             float* __restrict__ sums,
                             float* __restrict__ cnt,
                             int W, int D) {
    int per = D >> 2;
    long long total = (long long)W * per;
    long long stride = (long long)gridDim.x * blockDim.x;
    for (long long i = (long long)blockIdx.x * blockDim.x + threadIdx.x;
         i < total; i += stride) {
        int w  = (int)(i / per);
        int d4 = (int)(i % per) << 2;
        int item = item_ids[w];
        int word = word_ids[w];
        float4 v = *(const float4*)(word_emb + (size_t)word * D + d4);
        float* sp = sums + (size_t)item * D + d4;
        atomicAdd(sp + 0, v.x);
        atomicAdd(sp + 1, v.y);
        atomicAdd(sp + 2, v.z);
        atomicAdd(sp + 3, v.w);
        if (d4 == 0) atomicAdd(cnt + item, 1.0f);
    }
}

// sums -> sums / max(cnt,1)  (in place: becomes t_feat)
__global__ void tfeat_finalize(float* __restrict__ sums,
                               const float* __restrict__ cnt,
                               int I, int D) {
    int i = blockIdx.x * blockDim.x + threadIdx.x;
    if (i >= I * D) return;
    float c = cnt[i / D];
    sums[i] = sums[i] / fmaxf(c, 1.0f);
}

// ---------------------------------------------------------------------------
// WMMA f32 GEMM:  C[M x N] = act( A[M x K] @ W (+bias +add0 +add1) )
//   TRANSB==0 : W is K x N row-major      (x @ conv_w)
//   TRANSB==1 : W is N x K row-major      (x @ w.T)
// Grid: x = groups of 8 row-tiles (one per wave), y = 16-col tile (tn).
// All waves of a block share tn -> stage the 16 W columns once in LDS as
// sW[k][c].  TRANSB=0 staging uses gfx1250 async copy
// (global_load_async_to_lds_b128 + s_wait_asynccnt); TRANSB=1 stages with a
// register transpose.  Inner loop: 16x16x4 f32 WMMA, A from global (b64),
// B from LDS (conflict-free: lanes 0-15 banks k*16+r, lanes 16-31 +32).
// ---------------------------------------------------------------------------
template <int K, int TRANSB>
__global__ void wmma_gemm_f32_t(const float* __restrict__ A,
                                const float* __restrict__ W,
                                const float* __restrict__ bias,
                                const float* __restrict__ add0,
                                const float* __restrict__ add1,
                                float* __restrict__ C,
                                int M, int N, int act) {
    __shared__ float sW[K * 16];
    const int tn   = blockIdx.y << 4;
    const int lane = threadIdx.x & 31;
    const int wave = threadIdx.x >> 5;

    if (TRANSB) {
        // W is (N x K): column c of B tile is row (tn+c) of W (contiguous K).
        // Load float4 along k, transpose-store into sW[k][c].
        constexpr int CH = 16 * (K / 4);
        for (int ch = threadIdx.x; ch < CH; ch += blockDim.x) {
            int c  = ch / (K / 4);
            int kq = ch % (K / 4);
            float4 v = *(const float4*)(W + (size_t)(tn + c) * K + kq * 4);
            sW[(kq * 4 + 0) * 16 + c] = v.x;
            sW[(kq * 4 + 1) * 16 + c] = v.y;
            sW[(kq * 4 + 2) * 16 + c] = v.z;
            sW[(kq * 4 + 3) * 16 + c] = v.w;
        }
    } else {
        // W is (K x N): row k contributes 16 contiguous floats = 4 b128 chunks
        // -> async DMA straight into LDS, tracked by ASYNCcnt.
        constexpr int CH = K * 4;          // float4 chunks
        for (int ch = threadIdx.x; ch < CH; ch += blockDim.x) {
            int k    = ch >> 2;
            int part = ch & 3;
            const float* gsrc = W + (size_t)k * N + tn + part * 4;
            unsigned lds_off = (unsigned)(unsigned long long)
                               (lds_f32*)(sW + k * 16 + part * 4);
            unsigned long long ga = (unsigned long long)gsrc;
            asm volatile("global_load_async_to_lds_b128 %0, %1, off"
                         :: "v"(lds_off), "v"(ga) : "memory");
        }
        asm volatile("s_wait_asynccnt 0" ::: "memory");
    }
    __syncthreads();

    int tm = (blockIdx.x * (blockDim.x >> 5) + wave) << 4;
    if (tm >= M) return;

    const int r    = lane & 15;         // A row / C col within tile
    const int koff = (lane >> 4) << 1;  // lanes 0-15: K+0,1 ; lanes 16-31: K+2,3

    v8f acc = {};
    const float* arow = A + (size_t)(tm + r) * K;
#pragma unroll
    for (int k = 0; k < K; k += 4) {
        v2f a, b;
        a.x = arow[k + koff];
        a.y = arow[k + koff + 1];
        b.x = sW[(k + koff) * 16 + r];
        b.y = sW[(k + koff + 1) * 16 + r];
        acc = __builtin_amdgcn_wmma_f32_16x16x4_f32(
            false, a, false, b, (short)0, acc, false, false);
    }

    int rowBase = tm + ((lane >> 4) << 3);  // +8 for upper half-lanes
    int col = tn + r;
    float bv = bias ? bias[col] : 0.0f;
#pragma unroll
    for (int vr = 0; vr < 8; ++vr) {
        int row = rowBase + vr;
        size_t idx = (size_t)row * N + col;
        float v = acc[vr] + bv;
        if (add0) v += add0[idx];
        if (add1) v += add1[idx];
        if (act)  v = lrelu(v);
        C[idx] = v;
    }
}

template <int K, int TRANSB>
static void launch_gemm(const float* A, const float* W, const float* bias,
                        const float* add0, const float* add1, float* C,
                        int M, int N, int act, hipStream_t s) {
    dim3 grid((M / 16 + 7) / 8, N / 16);
    wmma_gemm_f32_t<K, TRANSB><<<grid, 256, 0, s>>>(A, W, bias, add0, add1,
                                                    C, M, N, act);
}

// ---------------------------------------------------------------------------
// Build + L2-normalize x rows: row<U reads pref, row>=U reads tf (already in x)
// ---------------------------------------------------------------------------
__global__ void build_norm_x(const float* __restrict__ pref,
                             float* __restrict__ x,
                             int U, int N, int D) {
    int row = blockIdx.x * (blockDim.x >> 5) + (threadIdx.x >> 5);
    if (row >= N) return;
    int lane = threadIdx.x & 31;
    const float* s = (row < U) ? (pref + (size_t)row * D)
                               : (x + (size_t)row * D);
    float4 v = *(const float4*)(s + lane * 4);
    float ss = v.x * v.x + v.y * v.y + v.z * v.z + v.w * v.w;
#pragma unroll
    for (int m = 16; m >= 1; m >>= 1) ss += __shfl_xor(ss, m, 32);
    float inv = 1.0f / fmaxf(sqrtf(ss), EPSN);
    float4 o = {v.x * inv, v.y * inv, v.z * inv, v.w * inv};
    *(float4*)(x + (size_t)row * D + lane * 4) = o;
}

// ---------------------------------------------------------------------------
// Edge scatter-add: h[dst[e]] += y[src[e]]   (float4 gather + f32 atomics)
// ---------------------------------------------------------------------------
__global__ void edge_scatter_add(const float* __restrict__ y,
                                 const int* __restrict__ src,
                                 const int* __restrict__ dst,
                                 float* __restrict__ h,
                                 int E, int D) {
    int per = D >> 2;
    long long total = (long long)E * per;
    long long stride = (long long)gridDim.x * blockDim.x;
    for (long long i = (long long)blockIdx.x * blockDim.x + threadIdx.x;
         i < total; i += stride) {
        int e  = (int)(i / per);
        int d4 = (int)(i % per) << 2;
        float4 v = *(const float4*)(y + (size_t)src[e] * D + d4);
        float* hp = h + (size_t)dst[e] * D + d4;
        atomicAdd(hp + 0, v.x);
        atomicAdd(hp + 1, v.y);
        atomicAdd(hp + 2, v.z);
        atomicAdd(hp + 3, v.w);
    }
}

__global__ void lrelu_inplace(float* __restrict__ p, long long n) {
    long long i = (long long)blockIdx.x * blockDim.x + threadIdx.x;
    long long stride = (long long)gridDim.x * blockDim.x;
    for (; i < n; i += stride) p[i] = lrelu(p[i]);
}

__global__ void accumulate(float* __restrict__ acc,
                           const float* __restrict__ x, long long n) {
    long long i = (long long)blockIdx.x * blockDim.x + threadIdx.x;
    long long stride = (long long)gridDim.x * blockDim.x;
    for (; i < n; i += stride) acc[i] += x[i];
}

// ---------------------------------------------------------------------------
// Scoring: pos_scores | neg_scores | pre_pos | pre_neg  (each B floats)
// post = acc/3  ->  post dot = acc_dot / 9
// ---------------------------------------------------------------------------
__global__ void score_kernel(const float* __restrict__ post_acc,
                             const float* __restrict__ t_rep,
                             const int* __restrict__ users,
                             const int* __restrict__ pos,
                             const int* __restrict__ neg,
                             float* __restrict__ out, int B, int D) {
    int b = blockIdx.x * blockDim.x + threadIdx.x;
    if (b >= B) return;
    const float* tu = t_rep + (size_t)users[b] * D;
    const float* tp = t_rep + (size_t)pos[b] * D;
    const float* tn = t_rep + (size_t)neg[b] * D;
    const float* pu = post_acc + (size_t)users[b] * D;
    const float* pp = post_acc + (size_t)pos[b] * D;
    const float* pn = post_acc + (size_t)neg[b] * D;
    float prep = 0.f, pren = 0.f, pop = 0.f, pon = 0.f;
    for (int d = 0; d < D; ++d) {
        prep += tu[d] * tp[d];
        pren += tu[d] * tn[d];
        pop  += pu[d] * pp[d];
        pon  += pu[d] * pn[d];
    }
    pop *= (1.0f / 9.0f);
    pon *= (1.0f / 9.0f);
    float sp = 1.0f / (1.0f + expf(-prep));
    float sn = 1.0f / (1.0f + expf(-pren));
    out[b]         = pop * sp;
    out[B + b]     = pon * sn;
    out[2 * B + b] = prep;
    out[3 * B + b] = pren;
}

// ---------------------------------------------------------------------------
// Host orchestration
// ---------------------------------------------------------------------------
extern "C" void kernel_launch(void* const* d_in, const int* in_sizes, int n_in,
                              void* d_out, int out_size, void* d_ws, size_t ws_size,
                              hipStream_t stream) {
    constexpr int U = 40000, I = 30000, N = U + I;
    constexpr int DL = 128, DX = 64;

    const float* v_feat  = (const float*)d_in[0];
    const float* a_feat  = (const float*)d_in[1];
    const float* word_emb= (const float*)d_in[2];
    const float* id_emb  = (const float*)d_in[3];
    const float* pref    = (const float*)d_in[4];
    const float* mlp_w   = (const float*)d_in[5];
    const float* mlp_b   = (const float*)d_in[6];
    const float* conv1_w = (const float*)d_in[7];
    const float* lin1_w  = (const float*)d_in[8];
    const float* lin1_b  = (const float*)d_in[9];
    const float* g1_w    = (const float*)d_in[10];
    const float* g1_b    = (const float*)d_in[11];
    const float* conv2_w = (const float*)d_in[12];
    const float* lin2_w  = (const float*)d_in[13];
    const float* lin2_b  = (const float*)d_in[14];
    const float* g2_w    = (const float*)d_in[15];
    const float* g2_b    = (const float*)d_in[16];
    const int*   edges   = (const int*)d_in[17];
    const int*   words   = (const int*)d_in[18];
    const int*   users   = (const int*)d_in[19];
    const int*   posn    = (const int*)d_in[20];
    const int*   negn    = (const int*)d_in[21];
    float* out = (float*)d_out;

    const int E = in_sizes[17] / 2;
    const int W = in_sizes[18] / 2;
    const int B = in_sizes[19];
    const int* src = edges;
    const int* dst = edges + E;
    const int* item_ids = words;
    const int* word_ids = words + W;

    // workspace partition (floats)
    float* ws = (float*)d_ws;
    size_t off = 0;
    float* x     = ws + off; off += (size_t)N * DL;   // normalized features
    float* y     = ws + off; off += (size_t)N * DL;   // conv messages
    float* h     = ws + off; off += (size_t)N * DL;   // scatter accumulator
    float* t1    = ws + off; off += (size_t)N * DX;   // lin path
    float* xs    = ws + off; off += (size_t)N * DX;   // layer output / final rep
    float* post  = ws + off; off += (size_t)N * DX;   // sum of reps
    float* tfeat = ws + off; off += (size_t)I * DL;   // word scatter sums -> t_feat
    float* cnt   = ws + off; off += (size_t)I;

    const int TB = 256;
    const int GS = 16384;  // grid-stride blocks for bandwidth kernels
    #define NB(n) ((int)(((n) + TB - 1) / TB))

    // ---- t_feat = scatter_mean(word_emb[word_ids], item_ids) ----
    zero_f32<<<GS, TB, 0, stream>>>(tfeat, (long long)I * DL);
    zero_f32<<<NB(I), TB, 0, stream>>>(cnt, (long long)I);
    word_scatter<<<GS, TB, 0, stream>>>(word_emb, item_ids, word_ids,
                                        tfeat, cnt, W, DL);
    tfeat_finalize<<<NB((long long)I * DL), TB, 0, stream>>>(tfeat, cnt, I, DL);

    zero_f32<<<GS, TB, 0, stream>>>(post, (long long)N * DX);

    const float* feats[3] = {v_feat, a_feat, tfeat};

    for (int m = 0; m < 3; ++m) {
        const float* prefm = pref + (size_t)m * U * DL;
        // tf = feat @ mlp_w[m].T + mlp_b[m]  -> items part of x
        launch_gemm<DL, 1>(feats[m], mlp_w + (size_t)m * DL * DL,
                           mlp_b + (size_t)m * DL, nullptr, nullptr,
                           x + (size_t)U * DL, I, DL, 0, stream);
        // x = normalize(concat(pref, tf))
        build_norm_x<<<(N + 7) / 8, TB, 0, stream>>>(prefm, x, U, N, DL);

        // ---- layer 1 (128 -> 64) ----
        launch_gemm<DL, 0>(x, conv1_w + (size_t)m * DL * DL, nullptr,
                           nullptr, nullptr, y, N, DL, 0, stream);
        zero_f32<<<GS, TB, 0, stream>>>(h, (long long)N * DL);
        edge_scatter_add<<<GS, TB, 0, stream>>>(y, src, dst, h, E, DL);
        lrelu_inplace<<<GS, TB, 0, stream>>>(h, (long long)N * DL);
        // t1 = lrelu(x @ lin1_w.T + lin1_b)
        launch_gemm<DL, 1>(x, lin1_w + (size_t)m * DX * DL,
                           lin1_b + (size_t)m * DX, nullptr, nullptr,
                           t1, N, DX, 1, stream);
        // xs = lrelu(h @ g1_w.T + g1_b + t1 + id_emb)
        launch_gemm<DL, 1>(h, g1_w + (size_t)m * DX * DL,
                           g1_b + (size_t)m * DX, t1, id_emb,
                           xs, N, DX, 1, stream);

        // ---- layer 2 (64 -> 64) ----
        launch_gemm<DX, 0>(xs, conv2_w + (size_t)m * DX * DX, nullptr,
                           nullptr, nullptr, y, N, DX, 0, stream);
        zero_f32<<<GS, TB, 0, stream>>>(h, (long long)N * DX);
        edge_scatter_add<<<GS, TB, 0, stream>>>(y, src, dst, h, E, DX);
        lrelu_inplace<<<GS, TB, 0, stream>>>(h, (long long)N * DX);
        launch_gemm<DX, 1>(xs, lin2_w + (size_t)m * DX * DX,
                           lin2_b + (size_t)m * DX, nullptr, nullptr,
                           t1, N, DX, 1, stream);
        launch_gemm<DX, 1>(h, g2_w + (size_t)m * DX * DX,
                           g2_b + (size_t)m * DX, t1, id_emb,
                           xs, N, DX, 1, stream);

        accumulate<<<GS, TB, 0, stream>>>(post, xs, (long long)N * DX);
        // after m==2, xs holds t_rep (pre_representation)
    }

    score_kernel<<<(B + TB - 1) / TB, TB, 0, stream>>>(
        post, xs, users, posn, negn, out, B, DX);
    #undef NB
}